// GATNet_d_37709812859546
// MI455X (gfx1250) — compile-verified
//
#include <hip/hip_runtime.h>
#include <hip/hip_bf16.h>

// ---------------------------------------------------------------------------
// GATNet forward for MI455X (gfx1250, wave32).
// GEMMs run on the f32 matrix pipe via V_WMMA_F32_16X16X4_F32 with 2x2
// register blocking per wave (32x32 C per wave, 128x64 per 8-wave block).
// ---------------------------------------------------------------------------

typedef float v2f __attribute__((ext_vector_type(2)));
typedef float v8f __attribute__((ext_vector_type(8)));

#define NEG_SLOPE 0.2f

// ---------------- float atomic max via IEEE integer ordering ----------------
__device__ __forceinline__ void atomicMaxF32(float* addr, float val) {
  if (val >= 0.0f) {
    atomicMax((int*)addr, __float_as_int(val));
  } else {
    atomicMin((unsigned int*)addr, __float_as_uint(val));
  }
}

// ---------------- generic f32 WMMA GEMM: C[M,N] = A[M,K] * B[K,N] -----------
// Block = 256 threads = 8 waves arranged 4 (rows) x 2 (cols); each wave owns a
// 32x32 C tile (2x2 of 16x16 WMMA tiles) -> block covers 128x64 of C.
// Fragment layout (ISA 7.12.2, 32-bit A 16x4 / B 4x16 / C 16x16):
//   A: lane<16 -> row=lane,    K=k0+0/1 ; lane>=16 -> row=lane-16, K=k0+2/3
//   B: lane<16 -> col=lane,    K=k0+0/1 ; lane>=16 -> col=lane-16, K=k0+2/3
//   C: vgpr i  -> row = base + i + (lane>=16 ? 8 : 0), col = base + (lane&15)
// A lane's A data only affects its own output row, and B data only its own
// output column, so out-of-range rows/cols use CLAMPED addresses (in-bounds
// loads, garbage masked at the store). Only the K tail needs value masking.
__global__ void gemm_wmma_f32(const float* __restrict__ A,
                              const float* __restrict__ B,
                              float* __restrict__ C,
                              int M, int K, int N) {
  const int lane    = threadIdx.x & 31;
  const int wave    = threadIdx.x >> 5;
  const int waveRow = wave >> 1;       // 0..3
  const int waveCol = wave & 1;        // 0..1
  const int rowBase = blockIdx.y * 128 + waveRow * 32;
  const int colBase = blockIdx.x * 64 + waveCol * 32;
  const int half    = lane >> 4;       // 0: K+0/1, 1: K+2/3
  const int l16     = lane & 15;
  const int m0      = rowBase + l16;        // A rows handled by this lane
  const int m1      = rowBase + 16 + l16;
  const int n0      = colBase + l16;        // B/C columns handled by this lane
  const int n1      = colBase + 16 + l16;
  const int m0c     = (m0 < M) ? m0 : (M - 1);   // clamped -> always in-bounds
  const int m1c     = (m1 < M) ? m1 : (M - 1);
  const int n0c     = (n0 < N) ? n0 : (N - 1);
  const int n1c     = (n1 < N) ? n1 : (N - 1);

  const float* ap0 = A + (long long)m0c * K + half * 2;      // +4 per iter
  const float* ap1 = A + (long long)m1c * K + half * 2;
  const float* bp0 = B + n0c + (long long)(half * 2) * N;    // +4N per iter
  const float* bp1 = B + n1c + (long long)(half * 2) * N;
  const long long bstep = (long long)4 * N;

  v8f acc00 = {}, acc01 = {}, acc10 = {}, acc11 = {};
  const int Kmain = K & ~3;
  int k0 = 0;
#pragma unroll 2
  for (; k0 < Kmain; k0 += 4) {
    v2f a0, a1, b0, b1;
    a0.x = ap0[0]; a0.y = ap0[1];     // contiguous pair -> global_load_b64
    a1.x = ap1[0]; a1.y = ap1[1];
    b0.x = bp0[0]; b0.y = bp0[N];
    b1.x = bp1[0]; b1.y = bp1[N];
    acc00 = __builtin_amdgcn_wmma_f32_16x16x4_f32(false, a0, false, b0, (short)0, acc00, false, false);
    acc01 = __builtin_amdgcn_wmma_f32_16x16x4_f32(false, a0, false, b1, (short)0, acc01, false, false);
    acc10 = __builtin_amdgcn_wmma_f32_16x16x4_f32(false, a1, false, b0, (short)0, acc10, false, false);
    acc11 = __builtin_amdgcn_wmma_f32_16x16x4_f32(false, a1, false, b1, (short)0, acc11, false, false);
    ap0 += 4; ap1 += 4; bp0 += bstep; bp1 += bstep;
  }
  if (k0 < K) {  // K % 4 in {1,2,3}; zero-mask out-of-range K (clamped loads)
    const int ka   = k0 + half * 2;
    const bool v0  = (ka < K);
    const bool v1  = (ka + 1 < K);
    const int kac0 = v0 ? ka : (K - 1);
    const int kac1 = v1 ? (ka + 1) : (K - 1);
    const float* ar0 = A + (long long)m0c * K;
    const float* ar1 = A + (long long)m1c * K;
    v2f a0, a1, b0, b1;
    a0.x = v0 ? ar0[kac0] : 0.0f;  a0.y = v1 ? ar0[kac1] : 0.0f;
    a1.x = v0 ? ar1[kac0] : 0.0f;  a1.y = v1 ? ar1[kac1] : 0.0f;
    b0.x = v0 ? B[(long long)kac0 * N + n0c] : 0.0f;
    b0.y = v1 ? B[(long long)kac1 * N + n0c] : 0.0f;
    b1.x = v0 ? B[(long long)kac0 * N + n1c] : 0.0f;
    b1.y = v1 ? B[(long long)kac1 * N + n1c] : 0.0f;
    acc00 = __builtin_amdgcn_wmma_f32_16x16x4_f32(false, a0, false, b0, (short)0, acc00, false, false);
    acc01 = __builtin_amdgcn_wmma_f32_16x16x4_f32(false, a0, false, b1, (short)0, acc01, false, false);
    acc10 = __builtin_amdgcn_wmma_f32_16x16x4_f32(false, a1, false, b0, (short)0, acc10, false, false);
    acc11 = __builtin_amdgcn_wmma_f32_16x16x4_f32(false, a1, false, b1, (short)0, acc11, false, false);
  }

#pragma unroll
  for (int i = 0; i < 8; ++i) {
    const int r0 = rowBase + i + half * 8;        // top tile rows
    const int r1 = r0 + 16;                       // bottom tile rows
    if (r0 < M) {
      if (n0 < N) C[(long long)r0 * N + n0] = acc00[i];
      if (n1 < N) C[(long long)r0 * N + n1] = acc01[i];
    }
    if (r1 < M) {
      if (n0 < N) C[(long long)r1 * N + n0] = acc10[i];
      if (n1 < N) C[(long long)r1 * N + n1] = acc11[i];
    }
  }
}

// ---------------- fill ------------------------------------------------------
__global__ void fill_f32(float* __restrict__ p, float v, long long n) {
  long long i = (long long)blockIdx.x * blockDim.x + threadIdx.x;
  if (i < n) p[i] = v;
}

// ---------------- attention logits: one wave per (node, head) ---------------
template <int H, int F>
__global__ void attn_logits(const float* __restrict__ h,
                            const float* __restrict__ a_src,
                            const float* __restrict__ a_dst,
                            float* __restrict__ alS, float* __restrict__ alD,
                            int Nn) {
  const int gw   = (int)(((long long)blockIdx.x * blockDim.x + threadIdx.x) >> 5);
  const int lane = threadIdx.x & 31;
  if (gw >= Nn * H) return;
  const int nIdx = gw / H;
  const int hIdx = gw - nIdx * H;
  const float* hp = h + (long long)nIdx * (H * F) + hIdx * F;
  const float* as = a_src + hIdx * F;
  const float* ad = a_dst + hIdx * F;
  float s = 0.0f, d = 0.0f;
  for (int f = lane; f < F; f += 32) {
    const float v = hp[f];
    s += v * as[f];
    d += v * ad[f];
  }
#pragma unroll
  for (int off = 16; off > 0; off >>= 1) {
    s += __shfl_xor(s, off, 32);
    d += __shfl_xor(d, off, 32);
  }
  if (lane == 0) { alS[gw] = s; alD[gw] = d; }
}

// ---------------- edge pass 1: leaky logits + segment max -------------------
template <int H>
__global__ void edge_logits_max(const int* __restrict__ srcE,
                                const int* __restrict__ dstE,
                                int E, int Nn,
                                const float* __restrict__ alS,
                                const float* __restrict__ alD,
                                float* __restrict__ eBuf,
                                float* __restrict__ mMax) {
  const int e = blockIdx.x * blockDim.x + threadIdx.x;
  const int Etot = E + Nn;
  if (e >= Etot) return;
  const int s = (e < E) ? srcE[e] : (e - E);   // self-loops appended
  const int d = (e < E) ? dstE[e] : (e - E);
#pragma unroll
  for (int h = 0; h < H; ++h) {
    float v = alS[s * H + h] + alD[d * H + h];
    v = (v > 0.0f) ? v : NEG_SLOPE * v;
    eBuf[(long long)e * H + h] = v;
    atomicMaxF32(&mMax[d * H + h], v);
  }
}

// ---------------- edge pass 2: exp(e - m[dst]) + segment sum ----------------
template <int H>
__global__ void edge_softmax_num(const int* __restrict__ srcE,
                                 const int* __restrict__ dstE,
                                 int E, int Nn,
                                 const float* __restrict__ mMax,
                                 float* __restrict__ eBuf,
                                 float* __restrict__ denom) {
  const int e = blockIdx.x * blockDim.x + threadIdx.x;
  const int Etot = E + Nn;
  if (e >= Etot) return;
  const int d = (e < E) ? dstE[e] : (e - E);
#pragma unroll
  for (int h = 0; h < H; ++h) {
    const float w = expf(eBuf[(long long)e * H + h] - mMax[d * H + h]);
    eBuf[(long long)e * H + h] = w;
    atomicAdd(&denom[d * H + h], w);
  }
}

// ---------------- edge pass 3: out[dst] += alpha * h[src] -------------------
template <int H, int F>
__global__ void edge_aggregate(const int* __restrict__ srcE,
                               const int* __restrict__ dstE,
                               int E, int Nn,
                               const float* __restrict__ h,
                               const float* __restrict__ eBuf,
                               const float* __restrict__ denom,
                               float* __restrict__ out) {
  const int e = blockIdx.x;              // one block per edge
  const int s = (e < E) ? srcE[e] : (e - E);
  const int d = (e < E) ? dstE[e] : (e - E);
  constexpr int HF = H * F;
  const float* hs = h + (long long)s * HF;
  float* od = out + (long long)d * HF;
  for (int f = threadIdx.x; f < HF; f += blockDim.x) {
    const int hh = f / F;                // constant divisor -> mul/shift
    const float alpha = eBuf[(long long)e * H + hh] / denom[d * H + hh];
    atomicAdd(&od[f], alpha * hs[f]);
  }
}

// ---------------- bias + ELU (in place) -------------------------------------
__global__ void bias_elu(float* __restrict__ x, const float* __restrict__ b,
                         long long n, int HF) {
  long long i = (long long)blockIdx.x * blockDim.x + threadIdx.x;
  if (i >= n) return;
  const float v = x[i] + b[(int)(i % HF)];
  x[i] = (v > 0.0f) ? v : (expf(v) - 1.0f);
}

// ---------------- bias + ReLU (in place) ------------------------------------
__global__ void bias_relu(float* __restrict__ x, const float* __restrict__ b,
                          long long n, int HF) {
  long long i = (long long)blockIdx.x * blockDim.x + threadIdx.x;
  if (i >= n) return;
  const float v = x[i] + b[(int)(i % HF)];
  x[i] = (v > 0.0f) ? v : 0.0f;
}

// ---------------- global max pool over batch --------------------------------
template <int F>
__global__ void pool_max(const float* __restrict__ xin,
                         const int* __restrict__ batch,
                         int Nn, float* __restrict__ g) {
  long long i = (long long)blockIdx.x * blockDim.x + threadIdx.x;
  if (i >= (long long)Nn * F) return;
  const int nIdx = (int)(i / F);
  const int f = (int)(i - (long long)nIdx * F);
  atomicMaxF32(&g[batch[nIdx] * F + f], xin[i]);
}

// ---------------- FC epilogue: out = relu(in + b) ---------------------------
__global__ void bias_relu_out(const float* __restrict__ in,
                              const float* __restrict__ b,
                              float* __restrict__ out, int n, int F) {
  int i = blockIdx.x * blockDim.x + threadIdx.x;
  if (i >= n) return;
  const float v = in[i] + b[i % F];
  out[i] = (v > 0.0f) ? v : 0.0f;
}

// ---------------------------------------------------------------------------
extern "C" void kernel_launch(void* const* d_in, const int* in_sizes, int n_in,
                              void* d_out, int out_size, void* d_ws, size_t ws_size,
                              hipStream_t stream) {
  (void)n_in; (void)ws_size;
  // setup_inputs order
  const float* x      = (const float*)d_in[0];   // [N,78]
  const int*   eidx   = (const int*)  d_in[1];   // [2,E]
  const int*   batch  = (const int*)  d_in[2];   // [N]
  const float* W1     = (const float*)d_in[3];   // [78,780]
  const float* a_src1 = (const float*)d_in[4];   // [10,78]
  const float* a_dst1 = (const float*)d_in[5];   // [10,78]
  const float* b1     = (const float*)d_in[6];   // [780]
  const float* W2     = (const float*)d_in[7];   // [780,128]
  const float* a_src2 = (const float*)d_in[8];   // [1,128]
  const float* a_dst2 = (const float*)d_in[9];   // [128]
  const float* b2     = (const float*)d_in[10];  // [128]
  const float* fc_w   = (const float*)d_in[11];  // [128,128]
  const float* fc_b   = (const float*)d_in[12];  // [128]
  float* out = (float*)d_out;                    // [256,128]

  const int N    = in_sizes[2];          // 50000
  const int E    = in_sizes[1] / 2;      // 300000
  const int Etot = E + N;                // with self-loops
  constexpr int FIN = 78, H1 = 10, FH1 = 78, HF1 = H1 * FH1;   // 780
  constexpr int FOUT = 128, NG = 256;

  const int* srcE = eidx;
  const int* dstE = eidx + E;

  // ---- workspace layout (floats) ----
  float* ws = (float*)d_ws;
  long long off = 0;
  float* H1BUF = ws + off; off += (long long)N * HF1;     // h1 [N,780]
  float* OUT1  = ws + off; off += (long long)N * HF1;     // agg1 -> elu [N,780]
  float* EBUF  = ws + off; off += (long long)Etot * H1;   // per-edge logits/weights
  float* ALS1  = ws + off; off += (long long)N * H1;
  float* ALD1  = ws + off; off += (long long)N * H1;
  float* M1    = ws + off; off += (long long)N * H1;
  float* DEN1  = ws + off; off += (long long)N * H1;
  float* ALS2  = ws + off; off += N;
  float* ALD2  = ws + off; off += N;
  float* M2    = ws + off; off += N;
  float* DEN2  = ws + off; off += N;
  float* G     = ws + off; off += NG * FOUT;
  float* FCT   = ws + off; off += NG * FOUT;
  // layer-2 big buffers reuse the H1 region (h1 dead after layer-1 aggregation)
  float* H2BUF = H1BUF;                                    // [N,128]
  float* OUT2  = H1BUF + (long long)N * FOUT;              // [N,128]
  float* EBUF2 = EBUF;                                     // [Etot]

  const float NEG_INF = -__builtin_huge_valf();
  const int TB = 256;
  auto blocksFor = [](long long n, int tb) { return (unsigned)((n + tb - 1) / tb); };

  // ============================ Layer 1 ====================================
  fill_f32<<<blocksFor((long long)N * HF1, TB), TB, 0, stream>>>(OUT1, 0.0f, (long long)N * HF1);
  fill_f32<<<blocksFor((long long)N * H1, TB), TB, 0, stream>>>(M1, NEG_INF, (long long)N * H1);
  fill_f32<<<blocksFor((long long)N * H1, TB), TB, 0, stream>>>(DEN1, 0.0f, (long long)N * H1);

  {
    dim3 grid((HF1 + 63) / 64, (N + 127) / 128);
    gemm_wmma_f32<<<grid, TB, 0, stream>>>(x, W1, H1BUF, N, FIN, HF1);
  }
  attn_logits<H1, FH1><<<blocksFor((long long)N * H1 * 32, TB), TB, 0, stream>>>(
      H1BUF, a_src1, a_dst1, ALS1, ALD1, N);
  edge_logits_max<H1><<<blocksFor(Etot, TB), TB, 0, stream>>>(
      srcE, dstE, E, N, ALS1, ALD1, EBUF, M1);
  edge_softmax_num<H1><<<blocksFor(Etot, TB), TB, 0, stream>>>(
      srcE, dstE, E, N, M1, EBUF, DEN1);
  edge_aggregate<H1, FH1><<<Etot, TB, 0, stream>>>(
      srcE, dstE, E, N, H1BUF, EBUF, DEN1, OUT1);
  bias_elu<<<blocksFor((long long)N * HF1, TB), TB, 0, stream>>>(
      OUT1, b1, (long long)N * HF1, HF1);

  // ============================ Layer 2 ====================================
  {
    dim3 grid((FOUT + 63) / 64, (N + 127) / 128);
    gemm_wmma_f32<<<grid, TB, 0, stream>>>(OUT1, W2, H2BUF, N, HF1, FOUT);
  }
  fill_f32<<<blocksFor((long long)N * FOUT, TB), TB, 0, stream>>>(OUT2, 0.0f, (long long)N * FOUT);
  fill_f32<<<blocksFor(N, TB), TB, 0, stream>>>(M2, NEG_INF, N);
  fill_f32<<<blocksFor(N, TB), TB, 0, stream>>>(DEN2, 0.0f, N);

  attn_logits<1, FOUT><<<blocksFor((long long)N * 32, TB), TB, 0, stream>>>(
      H2BUF, a_src2, a_dst2, ALS2, ALD2, N);
  edge_logits_max<1><<<blocksFor(Etot, TB), TB, 0, stream>>>(
      srcE, dstE, E, N, ALS2, ALD2, EBUF2, M2);
  edge_softmax_num<1><<<blocksFor(Etot, TB), TB, 0, stream>>>(
      srcE, dstE, E, N, M2, EBUF2, DEN2);
  edge_aggregate<1, FOUT><<<Etot, TB, 0, stream>>>(
      srcE, dstE, E, N, H2BUF, EBUF2, DEN2, OUT2);
  bias_relu<<<blocksFor((long long)N * FOUT, TB), TB, 0, stream>>>(
      OUT2, b2, (long long)N * FOUT, FOUT);

  // ===================== Pool + FC head ====================================
  fill_f32<<<blocksFor(NG * FOUT, TB), TB, 0, stream>>>(G, NEG_INF, NG * FOUT);
  pool_max<FOUT><<<blocksFor((long long)N * FOUT, TB), TB, 0, stream>>>(OUT2, batch, N, G);
  {
    dim3 grid((FOUT + 63) / 64, (NG + 127) / 128);
    gemm_wmma_f32<<<grid, TB, 0, stream>>>(G, fc_w, FCT, NG, FOUT, FOUT);
  }
  bias_relu_out<<<blocksFor(out_size, TB), TB, 0, stream>>>(FCT, fc_b, out, out_size, FOUT);
}